// PA_5592047419737
// MI455X (gfx1250) — compile-verified
//
#include <hip/hip_runtime.h>
#include <math.h>

typedef __attribute__((ext_vector_type(2))) float v2f;
typedef __attribute__((ext_vector_type(8))) float v8f;

#define BATCH   16
#define CH      256
#define NPLANE  16384   // H*W = 128*128
#define BN_EPS  1e-5f

// ---------------------------------------------------------------------------
// Kernel 1: per-(b,c) plane reduction -> mean, max, unbiased std.
// Sum and sum-of-squares are accumulated on the matrix pipe via
// V_WMMA_F32_16X16X4_F32 against an all-ones B matrix (row sums replicated
// across all 16 columns => summing every lane's 8 accumulators yields 16x the
// true sum). Max rides the VALU in parallel. Bandwidth-bound streaming pass.
// ---------------------------------------------------------------------------
__global__ __launch_bounds__(256) void pa_stats_kernel(
    const float* __restrict__ x, float* __restrict__ u) {
  const int bc = blockIdx.x;           // 0 .. B*C-1
  const int t  = threadIdx.x;          // 0 .. 255
  const float4* xp = (const float4*)(x + (size_t)bc * NPLANE);

  const v2f one2 = {1.f, 1.f};
  v8f s0 = {0.f,0.f,0.f,0.f,0.f,0.f,0.f,0.f};
  v8f s1 = s0, q0 = s0, q1 = s0;
  float mx = -3.402823466e38f;

#pragma unroll 4
  for (int i = 0; i < 16; ++i) {
    float4 v = xp[i * 256 + t];
    v2f a0 = {v.x, v.y};
    v2f a1 = {v.z, v.w};
    // sum accumulators (matrix pipe)
    s0 = __builtin_amdgcn_wmma_f32_16x16x4_f32(false, a0, false, one2,
                                               (short)0, s0, false, false);
    s1 = __builtin_amdgcn_wmma_f32_16x16x4_f32(false, a1, false, one2,
                                               (short)0, s1, false, false);
    // sum-of-squares accumulators
    v2f b0 = a0 * a0;
    v2f b1 = a1 * a1;
    q0 = __builtin_amdgcn_wmma_f32_16x16x4_f32(false, b0, false, one2,
                                               (short)0, q0, false, false);
    q1 = __builtin_amdgcn_wmma_f32_16x16x4_f32(false, b1, false, one2,
                                               (short)0, q1, false, false);
    // max on the VALU
    mx = fmaxf(mx, fmaxf(fmaxf(v.x, v.y), fmaxf(v.z, v.w)));
  }

  // Per-lane fold of the accumulator tile (16x over-counted; fixed below).
  float ps = 0.f, pq = 0.f;
#pragma unroll
  for (int i = 0; i < 8; ++i) { ps += s0[i] + s1[i]; pq += q0[i] + q1[i]; }

  __shared__ float rs[256], rq[256], rx[256];
  rs[t] = ps; rq[t] = pq; rx[t] = mx;
  __syncthreads();
  for (int sft = 128; sft > 0; sft >>= 1) {
    if (t < sft) {
      rs[t] += rs[t + sft];
      rq[t] += rq[t + sft];
      rx[t]  = fmaxf(rx[t], rx[t + sft]);
    }
    __syncthreads();
  }

  if (t == 0) {
    const float sum  = rs[0] * (1.f / 16.f);   // undo 16x WMMA replication
    const float ssq  = rq[0] * (1.f / 16.f);
    const float mean = sum * (1.f / (float)NPLANE);
    float var = (ssq - sum * mean) / (float)(NPLANE - 1);  // unbiased
    var = var > 0.f ? var : 0.f;
    u[bc * 3 + 0] = mean;
    u[bc * 3 + 1] = rx[0];
    u[bc * 3 + 2] = sqrtf(var);
  }
}

// ---------------------------------------------------------------------------
// Kernel 2: per-batch exact median over 768 stats (rank select, stable ties:
// matches sorted order, avg of order stats 383 & 384 == jnp.median for even
// count), then (u-med)^3, 3-tap dot with w, BN(eval), sigmoid -> g[b,c].
// ---------------------------------------------------------------------------
__global__ __launch_bounds__(256) void pa_gate_kernel(
    const float* __restrict__ u, const float* __restrict__ w,
    const float* __restrict__ gamma, const float* __restrict__ beta,
    const float* __restrict__ rmean, const float* __restrict__ rvar,
    float* __restrict__ g) {
  const int b = blockIdx.x;            // 0 .. BATCH-1
  const int t = threadIdx.x;           // 0 .. 255
  __shared__ float su[CH * 3];
  __shared__ float medv[2];

  su[t]       = u[b * CH * 3 + t];
  su[t + 256] = u[b * CH * 3 + t + 256];
  su[t + 512] = u[b * CH * 3 + t + 512];
  __syncthreads();

#pragma unroll
  for (int r = 0; r < 3; ++r) {
    const int i = t * 3 + r;
    const float v = su[i];
    int rank = 0;
    for (int j = 0; j < CH * 3; ++j) {
      const float uj = su[j];
      rank += (uj < v) || (uj == v && j < i);
    }
    if (rank == 383) medv[0] = v;
    if (rank == 384) medv[1] = v;
  }
  __syncthreads();

  const float med = 0.5f * (medv[0] + medv[1]);
  const int c = t;                     // one channel per thread (C == 256)
  float z = 0.f;
#pragma unroll
  for (int k = 0; k < 3; ++k) {
    const float d = su[c * 3 + k] - med;
    z += d * d * d * w[c * 3 + k];
  }
  z = (z - rmean[c]) * rsqrtf(rvar[c] + BN_EPS) * gamma[c] + beta[c];
  g[b * CH + c] = 1.f / (1.f + __expf(-z));
}

// ---------------------------------------------------------------------------
// Kernel 3: out = x * g[b,c] broadcast. Pure float4 stream (b128 load/store),
// 4 blocks per plane so g[bc] is block-uniform and L1/L2 resident.
// ---------------------------------------------------------------------------
__global__ __launch_bounds__(256) void pa_apply_kernel(
    const float* __restrict__ x, const float* __restrict__ g,
    float* __restrict__ out) {
  const int blk = blockIdx.x;          // 0 .. 16383 (4 blocks per plane)
  const int bc  = blk >> 2;
  const float gv = g[bc];
  const size_t base = (size_t)blk * 4096;
  const float4* xp = (const float4*)(x + base);
  float4* op = (float4*)(out + base);
#pragma unroll
  for (int j = 0; j < 4; ++j) {
    float4 v = xp[j * 256 + threadIdx.x];
    v.x *= gv; v.y *= gv; v.z *= gv; v.w *= gv;
    op[j * 256 + threadIdx.x] = v;
  }
}

extern "C" void kernel_launch(void* const* d_in, const int* in_sizes, int n_in,
                              void* d_out, int out_size, void* d_ws, size_t ws_size,
                              hipStream_t stream) {
  const float* x     = (const float*)d_in[0];  // [16,256,128,128]
  const float* w     = (const float*)d_in[1];  // [256,3]
  const float* gamma = (const float*)d_in[2];  // [256]
  const float* beta  = (const float*)d_in[3];  // [256]
  const float* rmean = (const float*)d_in[4];  // [256]
  const float* rvar  = (const float*)d_in[5];  // [256]
  float* out = (float*)d_out;                  // [16,256,128,128]

  float* u = (float*)d_ws;                     // [16*256, 3] stats
  float* g = u + BATCH * CH * 3;               // [16*256] gate

  pa_stats_kernel<<<BATCH * CH, 256, 0, stream>>>(x, u);
  pa_gate_kernel<<<BATCH, 256, 0, stream>>>(u, w, gamma, beta, rmean, rvar, g);
  pa_apply_kernel<<<BATCH * CH * 4, 256, 0, stream>>>(x, g, out);
}